// GATModel_29085518529144
// MI455X (gfx1250) — compile-verified
//
#include <hip/hip_runtime.h>
#include <hip/hip_bf16.h>
#include <math.h>

#define NNODES 50000
#define NEDGES 1600000
#define NETOT (NEDGES + NNODES)   // edges + self loops
#define HEADS 4
#define FDIM 128                  // heads * hide = heads * out
#define NEG_SLOPE 0.2f

typedef __attribute__((ext_vector_type(2))) float v2f;
typedef __attribute__((ext_vector_type(8))) float v8f;

// ---------------------------------------------------------------------------
// H[N x 128] = A[N x 128] @ W[128 x 128]  via V_WMMA_F32_16X16X4_F32
// One wave -> one 16x16 output tile. 8 waves/block -> 16-row x 128-col strip.
// A frag (16x4, MxK): lanes 0-15 hold K={0,1}, lanes 16-31 hold K={2,3}.
// B frag (4x16, KxN): mirrored K split, N = lane&15.
// C/D (16x16): VGPR r -> M=r (lanes<16) / M=r+8 (lanes>=16), N = lane&15.
// ---------------------------------------------------------------------------
__global__ __launch_bounds__(256) void gemm_wmma(const float* __restrict__ A,
                                                 const float* __restrict__ W,
                                                 float* __restrict__ H) {
  const int wave = threadIdx.x >> 5;   // 0..7 -> 16-col tile
  const int lane = threadIdx.x & 31;
  const int m0 = blockIdx.x * 16;
  const int n0 = wave * 16;
  const int l15 = lane & 15;
  const int hi = lane >> 4;            // 0 or 1 (lane half)
  const int koff = hi * 2;             // K offset within 4-step
  const float* __restrict__ arow = A + (m0 + l15) * FDIM;
  const float* __restrict__ wcol = W + (n0 + l15);
  v8f acc = {};
#pragma unroll 4
  for (int k = 0; k < FDIM; k += 4) {
    v2f a, b;
    a.x = arow[k + koff];
    a.y = arow[k + koff + 1];
    b.x = wcol[(k + koff) * FDIM];
    b.y = wcol[(k + koff + 1) * FDIM];
    acc = __builtin_amdgcn_wmma_f32_16x16x4_f32(
        /*neg_a=*/false, a, /*neg_b=*/false, b,
        /*c_mod=*/(short)0, acc, /*reuse_a=*/false, /*reuse_b=*/false);
  }
  float* __restrict__ hout = H + n0 + l15;
#pragma unroll
  for (int r = 0; r < 8; ++r)
    hout[(m0 + r + hi * 8) * FDIM] = acc[r];
}

// alpha_s[n,h] = sum_d H[n,h,d]*a_src[h,d] ; alpha_d likewise
__global__ void node_alpha(const float* __restrict__ H,
                           const float* __restrict__ a_src,
                           const float* __restrict__ a_dst,
                           float* __restrict__ AS, float* __restrict__ AD) {
  int i = blockIdx.x * blockDim.x + threadIdx.x;   // node*HEADS + h
  if (i >= NNODES * HEADS) return;
  int h = i & 3;
  const float* __restrict__ hp = H + (i >> 2) * FDIM + h * 32;
  const float* __restrict__ as = a_src + h * 32;
  const float* __restrict__ ad = a_dst + h * 32;
  float s = 0.f, d = 0.f;
#pragma unroll
  for (int j = 0; j < 32; ++j) { float v = hp[j]; s += v * as[j]; d += v * ad[j]; }
  AS[i] = s; AD[i] = d;
}

__global__ void init_buffers(float* __restrict__ AGG, float* __restrict__ EMAX,
                             float* __restrict__ DEN) {
  int i = blockIdx.x * blockDim.x + threadIdx.x;
  if (i < NNODES * FDIM) AGG[i] = 0.f;
  if (i < NNODES * HEADS) { EMAX[i] = -1e30f; DEN[i] = 0.f; }
}

__device__ __forceinline__ void atomic_max_f32(float* addr, float val) {
  // monotone bit trick: works across sign mix given -1e30 init
  if (val >= 0.f) atomicMax((int*)addr, __float_as_int(val));
  else            atomicMin((unsigned int*)addr, __float_as_uint(val));
}

__device__ __forceinline__ void edge_sd(int e, const int* __restrict__ esrc,
                                        const int* __restrict__ edst, int& s, int& d) {
  if (e < NEDGES) { s = esrc[e]; d = edst[e]; }
  else            { s = d = e - NEDGES; }          // self loop
}

__global__ void edge_logit_max(const int* __restrict__ esrc, const int* __restrict__ edst,
                               const float* __restrict__ AS, const float* __restrict__ AD,
                               float* __restrict__ E, float* __restrict__ EMAX) {
  long long i = (long long)blockIdx.x * blockDim.x + threadIdx.x;
  if (i >= (long long)NETOT * HEADS) return;
  int e = (int)(i >> 2), h = (int)(i & 3), s, d;
  edge_sd(e, esrc, edst, s, d);
  float v = AS[s * HEADS + h] + AD[d * HEADS + h];
  v = v > 0.f ? v : NEG_SLOPE * v;                 // leaky relu
  E[i] = v;
  atomic_max_f32(&EMAX[d * HEADS + h], v);
}

__global__ void edge_exp_sum(const int* __restrict__ edst, const float* __restrict__ EMAX,
                             float* __restrict__ E, float* __restrict__ DEN) {
  long long i = (long long)blockIdx.x * blockDim.x + threadIdx.x;
  if (i >= (long long)NETOT * HEADS) return;
  int e = (int)(i >> 2), h = (int)(i & 3);
  int d = (e < NEDGES) ? edst[e] : e - NEDGES;
  float ex = __expf(E[i] - EMAX[d * HEADS + h]);
  E[i] = ex;                                       // store exp in place
  atomicAdd(&DEN[d * HEADS + h], ex);
}

__global__ void edge_scatter(const int* __restrict__ esrc, const int* __restrict__ edst,
                             const float* __restrict__ E, const float* __restrict__ DEN,
                             const float* __restrict__ H, float* __restrict__ AGG) {
  long long i = (long long)blockIdx.x * blockDim.x + threadIdx.x;
  if (i >= (long long)NETOT * FDIM) return;
  int e = (int)(i >> 7), c = (int)(i & 127), h = c >> 5, s, d;
  edge_sd(e, esrc, edst, s, d);
  float alpha = E[(long long)e * HEADS + h] / (DEN[d * HEADS + h] + 1e-16f);
  atomicAdd(&AGG[d * FDIM + c], H[s * FDIM + c] * alpha);
}

__global__ void bias_relu(float* __restrict__ AGG, const float* __restrict__ b) {
  int i = blockIdx.x * blockDim.x + threadIdx.x;
  if (i >= NNODES * FDIM) return;
  float v = AGG[i] + b[i & 127];
  AGG[i] = v > 0.f ? v : 0.f;
}

__global__ void gather_out(const float* __restrict__ AGG, const float* __restrict__ b,
                           const int* __restrict__ root, float* __restrict__ out) {
  int i = blockIdx.x * blockDim.x + threadIdx.x;
  if (i >= 64 * FDIM) return;
  int r = i >> 7, c = i & 127;
  out[i] = AGG[root[r] * FDIM + c] + b[c];
}

extern "C" void kernel_launch(void* const* d_in, const int* in_sizes, int n_in,
                              void* d_out, int out_size, void* d_ws, size_t ws_size,
                              hipStream_t stream) {
  (void)in_sizes; (void)n_in; (void)out_size; (void)ws_size;
  const float* x      = (const float*)d_in[0];
  const int*   eidx   = (const int*)d_in[1];       // [2, NEDGES]
  const int*   root   = (const int*)d_in[2];       // [64]
  const float* W1     = (const float*)d_in[3];
  const float* a1_src = (const float*)d_in[4];
  const float* a1_dst = (const float*)d_in[5];
  const float* b1     = (const float*)d_in[6];
  const float* W2     = (const float*)d_in[7];
  const float* a2_src = (const float*)d_in[8];
  const float* a2_dst = (const float*)d_in[9];
  const float* b2     = (const float*)d_in[10];
  float* out = (float*)d_out;

  const int* esrc = eidx;
  const int* edst = eidx + NEDGES;

  // workspace layout (floats)
  float* H    = (float*)d_ws;                      // N*128
  float* AGG  = H + (size_t)NNODES * FDIM;         // N*128
  float* AS   = AGG + (size_t)NNODES * FDIM;       // N*4
  float* AD   = AS + NNODES * HEADS;
  float* EMAX = AD + NNODES * HEADS;
  float* DEN  = EMAX + NNODES * HEADS;
  float* E    = DEN + NNODES * HEADS;              // NETOT*4

  const int T = 256;
  const int gGemm = NNODES / 16;                                   // 3125
  const int gNA   = (NNODES * HEADS + T - 1) / T;
  const int gNF   = (NNODES * FDIM + T - 1) / T;
  const int gEH   = (int)(((long long)NETOT * HEADS + T - 1) / T);
  const int gEC   = (int)(((long long)NETOT * FDIM + T - 1) / T);

  // ---- layer 1 ----
  gemm_wmma<<<gGemm, T, 0, stream>>>(x, W1, H);
  node_alpha<<<gNA, T, 0, stream>>>(H, a1_src, a1_dst, AS, AD);
  init_buffers<<<gNF, T, 0, stream>>>(AGG, EMAX, DEN);
  edge_logit_max<<<gEH, T, 0, stream>>>(esrc, edst, AS, AD, E, EMAX);
  edge_exp_sum<<<gEH, T, 0, stream>>>(edst, EMAX, E, DEN);
  edge_scatter<<<gEC, T, 0, stream>>>(esrc, edst, E, DEN, H, AGG);
  bias_relu<<<gNF, T, 0, stream>>>(AGG, b1);

  // ---- layer 2 ----
  gemm_wmma<<<gGemm, T, 0, stream>>>(AGG, W2, H);
  node_alpha<<<gNA, T, 0, stream>>>(H, a2_src, a2_dst, AS, AD);
  init_buffers<<<gNF, T, 0, stream>>>(AGG, EMAX, DEN);
  edge_logit_max<<<gEH, T, 0, stream>>>(esrc, edst, AS, AD, E, EMAX);
  edge_exp_sum<<<gEH, T, 0, stream>>>(edst, EMAX, E, DEN);
  edge_scatter<<<gEC, T, 0, stream>>>(esrc, edst, E, DEN, H, AGG);

  gather_out<<<(64 * FDIM) / T, T, 0, stream>>>(AGG, b2, root, out);
}